// MacenkoStainNormalization_3332894622087
// MI455X (gfx1250) — compile-verified
//
#include <hip/hip_runtime.h>
#include <math.h>

// ---------------------------------------------------------------------------
// Macenko stain normalization for MI455X (gfx1250, wave32).
// Memory-bound: ~600MB min traffic -> ~26us at 23.3 TB/s. f32 throughout.
//
// WMMA (V_WMMA_F32_16X16X4_F32) computes the one real GEMM: cov = OD^T*OD,
// block-diagonal packed (5 groups x 4 pixels per instruction); the A register
// image doubles as B (=> D += A * A^T) under the ISA A/B layouts.
//
// L2 strategy: cov pass streams the 201MB input through the 192MB L2 with
// default (RT) loads; the apply pass stores its 201MB output NON-TEMPORALLY
// and reads the input with NT loads (last use), so input lines survive in L2
// between the two passes and apply's reads largely avoid HBM.
// ---------------------------------------------------------------------------

typedef __attribute__((ext_vector_type(2))) float v2f;
typedef __attribute__((ext_vector_type(4))) float v4f;
typedef __attribute__((ext_vector_type(8))) float v8f;

#define HW        (512 * 512)
#define N_IMGS    65                 // 64 batch images + 1 target
#define BETA_TH   0.15f
#define LN2_F     0.6931471805599453f
#define LN255_F   5.541263545158426f

// workspace layout (in floats)
#define WS_COV    0                  // 65 * 9
#define WS_W      1024               // 64 * 9

__device__ __forceinline__ float od_of(float x) {
    // -log((x+1)/255) = ln255 - ln2 * log2(x+1);  __log2f -> v_log_f32
    return fmaf(-__log2f(x + 1.0f), LN2_F, LN255_F);
}

// ---------------------------------------------------------------------------
// Kernel 0: zero the covariance accumulators (re-run every call; deterministic)
// ---------------------------------------------------------------------------
__global__ void mk_zero_cov(float* __restrict__ ws) {
    int t = blockIdx.x * blockDim.x + threadIdx.x;
    if (t < N_IMGS * 9) ws[WS_COV + t] = 0.0f;
}

// ---------------------------------------------------------------------------
// Kernel 1: per-image masked-OD covariance via WMMA f32 16x16x4.
//
// Tile packing: group g (0..4) lives in rows/cols 3g..3g+2 of the 16x16 D.
//   A[3g+i][k] = od_i(pixel base+4g+k)   (i = channel)
// ISA f32 A 16x4 layout: lane L<16 holds row L, K={0,1}; lane L+16 holds
// row L-16, K={2,3}. B 4x16 is the mirror, so passing the same v2f as A and
// B computes D += A * A^T; diagonal 3x3 blocks accumulate group covariances.
// grid = (blocksPerImage, 65), block = 256 (8 waves)
// ---------------------------------------------------------------------------
__device__ __forceinline__ v2f load2_guard(const float* __restrict__ p, int idx) {
    v2f r; r.x = 0.0f; r.y = 0.0f;
    if (idx + 1 < HW) {
        r = *(const v2f*)(p + idx);       // 8B aligned: idx is even
    } else if (idx < HW) {
        r.x = p[idx];
    }
    return r;
}

__global__ void __launch_bounds__(256)
mk_cov_wmma(const float* __restrict__ batch,
            const float* __restrict__ target,
            float* __restrict__ ws)
{
    const int b = blockIdx.y;                       // image 0..64
    const float* __restrict__ img =
        (b < 64) ? (batch + (size_t)b * 3u * HW) : target;

    const int lane = threadIdx.x & 31;
    const int warp = threadIdx.x >> 5;
    const int wavesPerBlock = blockDim.x >> 5;      // 8
    // scalarize the wave-uniform loop bounds (clean s_branch control flow,
    // EXEC provably all-ones at every WMMA)
    const int waveId = __builtin_amdgcn_readfirstlane(
                           blockIdx.x * wavesPerBlock + warp);
    const int nWaves = gridDim.x * wavesPerBlock;

    const int row = lane & 15;                      // tile row 0..15
    const int kk  = lane >> 4;                      // K-pair select (0 or 1)
    const int g   = row / 3;                        // pixel group 0..4
    const int ch  = row - 3 * g;                    // channel 0..2
    const bool active_row = (row < 15);             // row 15 unused
    const int laneOff = 4 * g + 2 * kk;             // lane's pixel pair offset

    v8f acc0 = {};                                  // 16x16 f32 accumulators
    v8f acc1 = {};

    const int nChunks = (HW + 19) / 20;             // 20 pixels per WMMA

    for (int c = waveId; c < nChunks; c += 2 * nWaves) {
        {   // ---- chunk c -> acc0 ----
            const int p = c * 20 + laneOff;
            v2f a; a.x = 0.0f; a.y = 0.0f;
            if (active_row) {
                v2f x0 = load2_guard(img + 0 * HW, p);
                v2f x1 = load2_guard(img + 1 * HW, p);
                v2f x2 = load2_guard(img + 2 * HW, p);
#pragma unroll
                for (int t = 0; t < 2; ++t) {
                    float od0 = od_of(t ? x0.y : x0.x);
                    float od1 = od_of(t ? x1.y : x1.x);
                    float od2 = od_of(t ? x2.y : x2.x);
                    bool keep = (od0 > BETA_TH) | (od1 > BETA_TH) | (od2 > BETA_TH);
                    bool inb  = (p + t) < HW;
                    float odc = (ch == 0) ? od0 : ((ch == 1) ? od1 : od2);
                    float val = (keep && inb) ? odc : 0.0f;
                    if (t == 0) a.x = val; else a.y = val;
                }
            }
            acc0 = __builtin_amdgcn_wmma_f32_16x16x4_f32(
                       false, a, false, a, (short)0, acc0, false, false);
        }
        const int c2 = c + nWaves;                  // scalar condition
        if (c2 < nChunks) {                         // ---- chunk c2 -> acc1 ----
            const int p = c2 * 20 + laneOff;
            v2f a; a.x = 0.0f; a.y = 0.0f;
            if (active_row) {
                v2f x0 = load2_guard(img + 0 * HW, p);
                v2f x1 = load2_guard(img + 1 * HW, p);
                v2f x2 = load2_guard(img + 2 * HW, p);
#pragma unroll
                for (int t = 0; t < 2; ++t) {
                    float od0 = od_of(t ? x0.y : x0.x);
                    float od1 = od_of(t ? x1.y : x1.x);
                    float od2 = od_of(t ? x2.y : x2.x);
                    bool keep = (od0 > BETA_TH) | (od1 > BETA_TH) | (od2 > BETA_TH);
                    bool inb  = (p + t) < HW;
                    float odc = (ch == 0) ? od0 : ((ch == 1) ? od1 : od2);
                    float val = (keep && inb) ? odc : 0.0f;
                    if (t == 0) a.x = val; else a.y = val;
                }
            }
            acc1 = __builtin_amdgcn_wmma_f32_16x16x4_f32(
                       false, a, false, a, (short)0, acc1, false, false);
        }
    }

    // Merge accumulators, dump D to LDS, fold 5 diagonal blocks + 8 waves.
    v8f acc = acc0 + acc1;
    __shared__ float lds[8][8][32];                 // wave x vgpr x lane
#pragma unroll
    for (int v = 0; v < 8; ++v) lds[warp][v][lane] = acc[v];
    __syncthreads();

    if (threadIdx.x < 9) {
        const int i = threadIdx.x / 3;
        const int j = threadIdx.x - 3 * i;
        float s = 0.0f;
        for (int w = 0; w < wavesPerBlock; ++w) {
#pragma unroll
            for (int gg = 0; gg < 5; ++gg) {
                const int r = 3 * gg + i;           // tile row
                const int cIdx = 3 * gg + j;        // tile col
                // D layout: VGPR v holds rows v (lanes 0-15) and v+8 (16-31)
                s += lds[w][r & 7][cIdx + ((r >> 3) << 4)];
            }
        }
        atomicAdd(&ws[WS_COV + b * 9 + threadIdx.x], s);
    }
}

// ---------------------------------------------------------------------------
// Kernel 2: 3x3 symmetric eigensolve (cyclic Jacobi) + Macenko postprocessing
// per image, then fuse W_b = inv(S_b^T) @ T^T.  One block, 96 threads.
// ---------------------------------------------------------------------------
__device__ void jacobi_rot(float A[3][3], float V[3][3], int p, int q) {
    const float apq = A[p][q];
    if (fabsf(apq) < 1e-20f) return;
    const float theta = (A[q][q] - A[p][p]) / (2.0f * apq);
    float tt = 1.0f / (fabsf(theta) + sqrtf(theta * theta + 1.0f));
    if (theta < 0.0f) tt = -tt;
    const float cc = 1.0f / sqrtf(tt * tt + 1.0f);
    const float ss = tt * cc;
    const float app = A[p][p], aqq = A[q][q];
    A[p][p] = cc * cc * app - 2.0f * ss * cc * apq + ss * ss * aqq;
    A[q][q] = ss * ss * app + 2.0f * ss * cc * apq + cc * cc * aqq;
    A[p][q] = A[q][p] = 0.0f;
    const int r = 3 - p - q;                        // remaining index
    const float arp = A[r][p], arq = A[r][q];
    A[r][p] = A[p][r] = cc * arp - ss * arq;
    A[r][q] = A[q][r] = ss * arp + cc * arq;
    for (int i = 0; i < 3; ++i) {
        const float vip = V[i][p], viq = V[i][q];
        V[i][p] = cc * vip - ss * viq;
        V[i][q] = ss * vip + cc * viq;
    }
}

__global__ void mk_solve(float* __restrict__ ws) {
    __shared__ float Tsh[9];                        // target stain matrix
    const int t = threadIdx.x;
    float S[3][3];

    if (t < N_IMGS) {
        const float* c = ws + WS_COV + t * 9;
        float A[3][3] = {{c[0], c[1], c[2]}, {c[3], c[4], c[5]}, {c[6], c[7], c[8]}};
        // symmetrize (block-diagonal WMMA packing + atomic order)
        float a01 = 0.5f * (A[0][1] + A[1][0]);
        float a02 = 0.5f * (A[0][2] + A[2][0]);
        float a12 = 0.5f * (A[1][2] + A[2][1]);
        A[0][1] = A[1][0] = a01; A[0][2] = A[2][0] = a02; A[1][2] = A[2][1] = a12;

        float V[3][3] = {{1, 0, 0}, {0, 1, 0}, {0, 0, 1}};
        for (int sweep = 0; sweep < 12; ++sweep) {
            jacobi_rot(A, V, 0, 1);
            jacobi_rot(A, V, 0, 2);
            jacobi_rot(A, V, 1, 2);
        }

        // sort columns by first-row value, descending (argsort(-eigvecs[0]))
        for (int a = 0; a < 2; ++a)
            for (int b2 = a + 1; b2 < 3; ++b2)
                if (V[0][a] < V[0][b2])
                    for (int r = 0; r < 3; ++r) {
                        float tmp = V[r][a]; V[r][a] = V[r][b2]; V[r][b2] = tmp;
                    }
        // row-normalize (reference: norm over axis=1)
        for (int r = 0; r < 3; ++r) {
            float n = 1.0f / sqrtf(V[r][0] * V[r][0] + V[r][1] * V[r][1] +
                                   V[r][2] * V[r][2]);
            V[r][0] *= n; V[r][1] *= n; V[r][2] *= n;
        }
        // sign fix: cols 0,1 by first-row sign; col 2 forced +1
        float sg0 = (V[0][0] < 0.0f) ? -1.0f : 1.0f;
        float sg1 = (V[0][1] < 0.0f) ? -1.0f : 1.0f;
        for (int r = 0; r < 3; ++r) { V[r][0] *= sg0; V[r][1] *= sg1; }
        // diag([ALPHA,1,1]) with ALPHA=1 -> identity
        for (int r = 0; r < 3; ++r)
            for (int cI = 0; cI < 3; ++cI) S[r][cI] = V[r][cI];

        if (t == 64)
            for (int k = 0; k < 9; ++k) Tsh[k] = S[k / 3][k % 3];
    }
    __syncthreads();

    if (t < 64) {
        // inv(S) via adjugate; inv(S^T) = inv(S)^T
        float d00 = S[1][1] * S[2][2] - S[1][2] * S[2][1];
        float d01 = S[1][0] * S[2][2] - S[1][2] * S[2][0];
        float d02 = S[1][0] * S[2][1] - S[1][1] * S[2][0];
        float invDet = 1.0f / (S[0][0] * d00 - S[0][1] * d01 + S[0][2] * d02);
        float iS[3][3];
        iS[0][0] =  d00 * invDet;
        iS[0][1] = -(S[0][1] * S[2][2] - S[0][2] * S[2][1]) * invDet;
        iS[0][2] =  (S[0][1] * S[1][2] - S[0][2] * S[1][1]) * invDet;
        iS[1][0] = -d01 * invDet;
        iS[1][1] =  (S[0][0] * S[2][2] - S[0][2] * S[2][0]) * invDet;
        iS[1][2] = -(S[0][0] * S[1][2] - S[0][2] * S[1][0]) * invDet;
        iS[2][0] =  d02 * invDet;
        iS[2][1] = -(S[0][0] * S[2][1] - S[0][1] * S[2][0]) * invDet;
        iS[2][2] =  (S[0][0] * S[1][1] - S[0][1] * S[1][0]) * invDet;
        // W[c][e] = sum_d inv(S^T)[c][d] * T[e][d] = sum_d iS[d][c] * T[e][d]
        for (int cI = 0; cI < 3; ++cI)
            for (int e = 0; e < 3; ++e) {
                float acc = 0.0f;
                for (int d = 0; d < 3; ++d) acc += iS[d][cI] * Tsh[e * 3 + d];
                ws[WS_W + t * 9 + cI * 3 + e] = acc;
            }
    }
}

// ---------------------------------------------------------------------------
// Kernel 3: per-pixel normalization, float4-vectorized, fully coalesced.
// NT loads (last use of input) + NT stores (keep input resident in L2).
// grid = (blocksPerImage, 64), block = 256
// ---------------------------------------------------------------------------
__global__ void __launch_bounds__(256)
mk_apply(const float* __restrict__ batch,
         const float* __restrict__ ws,
         float* __restrict__ out)
{
    const int b = blockIdx.y;
    const float* __restrict__ Wp = ws + WS_W + b * 9;   // uniform -> s_loads
    const float w00 = Wp[0], w01 = Wp[1], w02 = Wp[2];
    const float w10 = Wp[3], w11 = Wp[4], w12 = Wp[5];
    const float w20 = Wp[6], w21 = Wp[7], w22 = Wp[8];

    const float* __restrict__ in = batch + (size_t)b * 3u * HW;
    float* __restrict__ o = out + (size_t)b * 3u * HW;
    const v4f* __restrict__ p0 = (const v4f*)(in);
    const v4f* __restrict__ p1 = (const v4f*)(in + HW);
    const v4f* __restrict__ p2 = (const v4f*)(in + 2 * HW);
    v4f* __restrict__ q0 = (v4f*)(o);
    v4f* __restrict__ q1 = (v4f*)(o + HW);
    v4f* __restrict__ q2 = (v4f*)(o + 2 * HW);

    const int nvec = HW / 4;
    const int stride = gridDim.x * blockDim.x;
    for (int i = blockIdx.x * blockDim.x + threadIdx.x; i < nvec; i += stride) {
        const int nxt = i + stride;                 // prefetch next iteration
        if (nxt < nvec) {
            __builtin_prefetch(p0 + nxt, 0, 3);
            __builtin_prefetch(p1 + nxt, 0, 3);
            __builtin_prefetch(p2 + nxt, 0, 3);
        }
        v4f x0 = __builtin_nontemporal_load(p0 + i);
        v4f x1 = __builtin_nontemporal_load(p1 + i);
        v4f x2 = __builtin_nontemporal_load(p2 + i);
        v4f r0, r1, r2;
#pragma unroll
        for (int l = 0; l < 4; ++l) {
            const float od0 = od_of(x0[l]);
            const float od1 = od_of(x1[l]);
            const float od2 = od_of(x2[l]);
            const float t0 = od0 * w00 + od1 * w10 + od2 * w20;
            const float t1 = od0 * w01 + od1 * w11 + od2 * w21;
            const float t2 = od0 * w02 + od1 * w12 + od2 * w22;
            r0[l] = fminf(fmaxf(255.0f * __expf(-t0), 0.0f), 255.0f);
            r1[l] = fminf(fmaxf(255.0f * __expf(-t1), 0.0f), 255.0f);
            r2[l] = fminf(fmaxf(255.0f * __expf(-t2), 0.0f), 255.0f);
        }
        __builtin_nontemporal_store(r0, q0 + i);
        __builtin_nontemporal_store(r1, q1 + i);
        __builtin_nontemporal_store(r2, q2 + i);
    }
}

// ---------------------------------------------------------------------------
extern "C" void kernel_launch(void* const* d_in, const int* in_sizes, int n_in,
                              void* d_out, int out_size, void* d_ws, size_t ws_size,
                              hipStream_t stream) {
    (void)in_sizes; (void)n_in; (void)out_size; (void)ws_size;
    const float* batch  = (const float*)d_in[0];   // (64,3,512,512) f32
    const float* target = (const float*)d_in[1];   // (3,512,512) f32
    float* out = (float*)d_out;                    // (64,3,512,512) f32
    float* ws  = (float*)d_ws;

    mk_zero_cov<<<dim3(3), dim3(256), 0, stream>>>(ws);
    mk_cov_wmma<<<dim3(16, N_IMGS), dim3(256), 0, stream>>>(batch, target, ws);
    mk_solve<<<dim3(1), dim3(96), 0, stream>>>(ws);
    mk_apply<<<dim3(64, 64), dim3(256), 0, stream>>>(batch, ws, out);
}